// GlobalAttention_jittable_48017734369445
// MI455X (gfx1250) — compile-verified
//
#include <hip/hip_runtime.h>
#include <hip/hip_bf16.h>
#include <math.h>

#define GSEG 1024
#define DDIM 256
#define TILE 32

typedef __attribute__((ext_vector_type(2))) float v2f;
typedef __attribute__((ext_vector_type(4))) float f4v;
typedef __attribute__((ext_vector_type(8))) float v8f;
typedef __attribute__((ext_vector_type(4))) int   v4i;

#if __has_builtin(__builtin_amdgcn_global_load_async_to_lds_b128) && \
    __has_builtin(__builtin_amdgcn_s_wait_asynccnt)
#define ASYNC_TILE 1
#else
#define ASYNC_TILE 0
#endif

// ---------------------------------------------------------------------------
// Kernel 1: segment row ranges from sorted batch ids.
// starts[s] = lower_bound(batch, s); starts[GSEG] = N.
// ---------------------------------------------------------------------------
__global__ void seg_ranges_kernel(const int* __restrict__ batch, int n,
                                  int* __restrict__ starts) {
    int s = blockIdx.x * blockDim.x + threadIdx.x;
    if (s > GSEG) return;
    int lo = 0, hi = n;
    while (lo < hi) {
        int mid = (lo + hi) >> 1;
        if (batch[mid] < s) lo = mid + 1; else hi = mid;
    }
    starts[s] = lo;
}

// ---------------------------------------------------------------------------
// Tile feed: issue one 32x256 f32 tile into an LDS buffer.
// Async path: GLOBAL_LOAD_ASYNC_TO_LDS_B128 per lane (EXEC-masked, ASYNCcnt),
// rows past the segment end are zero-filled with predicated ds_stores.
// ---------------------------------------------------------------------------
__device__ __forceinline__ void issue_tile(const float* __restrict__ x,
                                           float* __restrict__ buf,
                                           int base, int r1, int tid) {
    #pragma unroll
    for (int i = 0; i < 8; ++i) {
        const int f  = i * 256 + tid;   // float4 index within tile
        const int rl = f >> 6;          // tile row
        const int c4 = f & 63;          // float4 column
        const int rg = base + rl;
        float* lp = buf + rl * DDIM + c4 * 4;
        if (rg < r1) {
#if ASYNC_TILE
            __builtin_amdgcn_global_load_async_to_lds_b128(
                (__attribute__((address_space(1))) v4i*)(x + (size_t)rg * DDIM + c4 * 4),
                (__attribute__((address_space(3))) v4i*)lp,
                0, 0);
#else
            *(f4v*)lp = *(const f4v*)(x + (size_t)rg * DDIM + c4 * 4);
#endif
        } else {
            f4v z = {};
            *(f4v*)lp = z;
        }
    }
}

__device__ __forceinline__ void wait_tile() {
#if ASYNC_TILE
    __builtin_amdgcn_s_wait_asynccnt(0);
#endif
}

// ---------------------------------------------------------------------------
// Kernel 2: one workgroup per segment, single pass over x (online softmax),
// double-buffered async LDS tile feed, WMMA f32 16x16x4 rank-update.
// ---------------------------------------------------------------------------
__launch_bounds__(256)
__global__ void seg_softmax_attn_kernel(const float* __restrict__ x,
                                        const float* __restrict__ Wg,
                                        const float* __restrict__ bg,
                                        const int* __restrict__ starts,
                                        float* __restrict__ out) {
    __shared__ __align__(16) float xs[2][TILE * DDIM];  // 2 x 32 KB tiles
    __shared__ float gls[TILE];                         // gate values of tile
    __shared__ float els[TILE];                         // exp(g - m) of tile

    const int tid  = threadIdx.x;
    const int lane = tid & 31;
    const int wave = tid >> 5;          // 0..7
    const int seg  = blockIdx.x;
    const int r0   = starts[seg];
    const int r1   = starts[seg + 1];
    const float bias = bg[0];

    // Per-lane slice of gate weights: cols [4*lane,+4) and [128+4*lane,+4).
    const f4v w0 = *(const f4v*)(Wg + 4 * lane);
    const f4v w1 = *(const f4v*)(Wg + 128 + 4 * lane);

    // Wave owns output column blocks cb0 = 2*wave, cb1 = 2*wave+1.
    const int cb0 = 2 * wave, cb1 = 2 * wave + 1;
    v8f c0 = {};  // WMMA accumulators; all 16 M-rows identical by construction
    v8f c1 = {};
    float m = -3.0e38f;  // running max
    float s = 0.0f;      // running sum of exp

    const int ntiles = (r1 - r0 + TILE - 1) / TILE;

    if (ntiles > 0) issue_tile(x, xs[0], r0, r1, tid);   // prime the pipeline

    for (int t = 0; t < ntiles; ++t) {
        float* cur = xs[t & 1];

        wait_tile();        // this wave's async transfers for tile t arrived
        __syncthreads();    // tile t fully in LDS; tile (t+1)&1 reads retired

        // kick off tile t+1 into the other buffer (overlaps all compute below)
        if (t + 1 < ntiles) issue_tile(x, xs[(t + 1) & 1], r0 + (t + 1) * TILE, r1, tid);

        // ---- gate: wave w computes rows w, w+8, w+16, w+24 of the tile
        const int base = r0 + t * TILE;
        #pragma unroll
        for (int j = 0; j < 4; ++j) {
            const int rl = wave + 8 * j;
            const f4v a0 = *(const f4v*)(cur + rl * DDIM + 4 * lane);
            const f4v a1 = *(const f4v*)(cur + rl * DDIM + 128 + 4 * lane);
            float p = a0.x * w0.x + a0.y * w0.y + a0.z * w0.z + a0.w * w0.w
                    + a1.x * w1.x + a1.y * w1.y + a1.z * w1.z + a1.w * w1.w;
            #pragma unroll
            for (int k = 16; k >= 1; k >>= 1) p += __shfl_xor(p, k, 32);
            if (lane == 0) {
                const int rg = base + rl;
                gls[rl] = (rg < r1) ? (p + bias) : -3.0e38f;  // pad rows -> -inf
            }
        }
        __syncthreads();

        // ---- online softmax update (redundant & identical across waves)
        const float gl = gls[lane];
        float tmax = gl;
        #pragma unroll
        for (int k = 16; k >= 1; k >>= 1) tmax = fmaxf(tmax, __shfl_xor(tmax, k, 32));
        const float mn    = fmaxf(m, tmax);
        const float scale = __expf(m - mn);      // 0 on first tile (m = -inf)
        const float el    = __expf(gl - mn);     // pad rows -> exp(-huge) = 0
        float es = el;
        #pragma unroll
        for (int k = 16; k >= 1; k >>= 1) es += __shfl_xor(es, k, 32);
        s = s * scale + es;
        m = mn;
        c0 *= scale;
        c1 *= scale;
        if (wave == 0) els[lane] = el;
        __syncthreads();

        // ---- WMMA rank-update: C(16x16) += A(16x4) * B(4x16), K = tile rows.
        // A[m][k] = e[kc+k] for every m (replicated rows); B[k][n] = x tile.
#if __has_builtin(__builtin_amdgcn_wmma_f32_16x16x4_f32)
        #pragma unroll
        for (int kc = 0; kc < TILE; kc += 4) {
            v2f a;
            // A 16x4 layout: v0 = K0 (lanes 0-15) / K2 (lanes 16-31); v1 = K1/K3
            a.x = els[kc + ((lane < 16) ? 0 : 2)];
            a.y = els[kc + ((lane < 16) ? 1 : 3)];
            // B 4x16 layout (assumed): v0 = row K0 (lanes 0-15) / K1 (16-31);
            //                          v1 = row K2 / K3
            const int krA = kc + ((lane < 16) ? 0 : 1);
            const int krB = kc + ((lane < 16) ? 2 : 3);
            const int n   = lane & 15;
            v2f b0, b1;
            b0.x = cur[krA * DDIM + cb0 * 16 + n];
            b0.y = cur[krB * DDIM + cb0 * 16 + n];
            b1.x = cur[krA * DDIM + cb1 * 16 + n];
            b1.y = cur[krB * DDIM + cb1 * 16 + n];
            c0 = __builtin_amdgcn_wmma_f32_16x16x4_f32(
                     false, a, false, b0, (short)0, c0, false, false);
            c1 = __builtin_amdgcn_wmma_f32_16x16x4_f32(
                     false, a, false, b1, (short)0, c1, false, false);
        }
#else
        // VALU fallback: accumulate into c0[0]/c1[0] only (lanes 0-15 valid)
        if (lane < 16) {
            #pragma unroll 4
            for (int r = 0; r < TILE; ++r) {
                const float ev = els[r];
                c0[0] += ev * cur[r * DDIM + cb0 * 16 + lane];
                c1[0] += ev * cur[r * DDIM + cb1 * 16 + lane];
            }
        }
#endif
    }

    // ---- epilogue: out[seg] = acc / (s + 1e-16). C rows identical; take
    // row 0 = VGPR 0, lanes 0-15 (N = 0..15).
    const float inv = 1.0f / (s + 1e-16f);
    if (lane < 16) {
        out[(size_t)seg * DDIM + cb0 * 16 + lane] = c0[0] * inv;
        out[(size_t)seg * DDIM + cb1 * 16 + lane] = c1[0] * inv;
    }
}

// ---------------------------------------------------------------------------
// Launch
// ---------------------------------------------------------------------------
extern "C" void kernel_launch(void* const* d_in, const int* in_sizes, int n_in,
                              void* d_out, int out_size, void* d_ws, size_t ws_size,
                              hipStream_t stream) {
    const float* x     = (const float*)d_in[0];  // (N, 256)
    const int*   batch = (const int*)d_in[1];    // (N,) sorted
    const float* Wg    = (const float*)d_in[2];  // (256, 1)
    const float* bg    = (const float*)d_in[3];  // (1,)
    float* out = (float*)d_out;                  // (GSEG, 256)
    const int n = in_sizes[0] / DDIM;            // N rows

    int* starts = (int*)d_ws;                    // (GSEG+1) ints in scratch

    seg_ranges_kernel<<<(GSEG + 1 + 255) / 256, 256, 0, stream>>>(batch, n, starts);
    seg_softmax_attn_kernel<<<GSEG, 256, 0, stream>>>(x, Wg, bg, starts, out);
}